// FeatureInteractionModel_61134564491862
// MI455X (gfx1250) — compile-verified
//
#include <hip/hip_runtime.h>

// Problem constants (from the reference).
#define PDIM 512
#define NROWS 200000
#define WAVES_PER_BLOCK 8
#define ROWS_PER_WAVE 16
#define ROWS_PER_BLOCK (WAVES_PER_BLOCK * ROWS_PER_WAVE)   // 128
#define XL_STRIDE 516           // padded X-tile row stride (floats) -> conflict-free ds_load_2addr_b64
#define NT_TILES (PDIM / 16)    // 32 column tiles
#define HALF_STEPS (NT_TILES * 2)   // 64 pipeline steps (tile x K-half)
#define HK_CHUNKS 64            // WMMA K-chunks per half step (256/4)
#define TH_TILE_FLOATS 8192     // one 512x16 Theta tile (pair-swizzled)
#define TH_HALF_FLOATS 4096     // one K-half of a tile (16 KB)

typedef __attribute__((ext_vector_type(2))) float v2f;
typedef __attribute__((ext_vector_type(8))) float v8f;
typedef __attribute__((ext_vector_type(4))) int   v4i;

#if __has_builtin(__builtin_amdgcn_global_load_async_to_lds_b128)
#define HAVE_ASYNC_LDS 1
// Builtin signature (from hipcc diagnostic): arg0 = v4i in addrspace(1) [printed
// as "__device__"], arg1 = v4i in addrspace(3), then int offset, int cpol.
typedef __attribute__((address_space(1))) v4i* async_gptr;
typedef __attribute__((address_space(3))) v4i* async_lptr;
#else
#define HAVE_ASYNC_LDS 0
#endif

// ---------------------------------------------------------------------------
// Kernel 1: scatter packed strictly-upper theta DIRECTLY into the N-tile-major,
// K-pair-swizzled layout the GEMM consumes:
//   ThetaS[(n>>4)*8192 + (k>>1)*32 + (n&15)*2 + (k&1)] = Theta[k][n]
// K-halves of a tile are contiguous 16 KB blocks.
// ---------------------------------------------------------------------------
__global__ void fim_build_theta(const float* __restrict__ tu,
                                float* __restrict__ ThetaS) {
    int idx = blockIdx.x * 256 + threadIdx.x;      // 512*512 total
    int a = idx >> 9;          // contraction index (Theta row)
    int b = idx & 511;         // output column
    float v = 0.0f;
    if (b > a) {
        int base = a * (PDIM - 1) - (a * (a - 1)) / 2;
        v = tu[base + (b - a - 1)];
    }
    int dst = (b >> 4) * TH_TILE_FLOATS + (a >> 1) * 32 + (b & 15) * 2 + (a & 1);
    ThetaS[dst] = v;
}

// Stage one 16 KB Theta half-tile: async global->LDS (no VGPR liveness),
// or a plain copy if the async builtin is unavailable.
__device__ __forceinline__ void stage_half(float* dst, const float* src, int tid) {
#if HAVE_ASYNC_LDS
    #pragma unroll
    for (int q = 0; q < 4; ++q) {
        int f = (q * 256 + tid) * 4;     // float index, 16B granules
        __builtin_amdgcn_global_load_async_to_lds_b128(
            (async_gptr)(src + f), (async_lptr)(dst + f),
            /*offset=*/0, /*cpol=*/0);
    }
#else
    #pragma unroll
    for (int q = 0; q < 4; ++q) {
        int f = (q * 256 + tid) * 4;
        float4 v = *(const float4*)(src + f);
        *(float4*)(dst + f) = v;
    }
#endif
}

// ---------------------------------------------------------------------------
// Kernel 2: fused  out[i] = beta0 + sum_n (C[i][n] + beta[n]) * X[i][n]
// where C = X @ Theta via V_WMMA_F32_16X16X4_F32.
// 8 wave32s; each wave owns a 16-row X tile. Theta half-tiles are
// double-buffered in LDS and filled by async global->LDS loads that overlap
// the WMMA phase (tracked with ASYNCcnt).
// ---------------------------------------------------------------------------
__launch_bounds__(256)
__global__ void fim_main(const float* __restrict__ X,
                         const float* __restrict__ beta0p,
                         const float* __restrict__ beta,
                         const float* __restrict__ ThetaS,
                         float* __restrict__ out) {
    extern __shared__ float smem[];
    // smem[0..4095]    : Theta half-tile buffer 0
    // smem[4096..8191] : Theta half-tile buffer 1
    float* xl = smem + 2 * TH_HALF_FLOATS;   // 8 waves * 16 rows * XL_STRIDE

    const int tid  = threadIdx.x;
    const int lane = tid & 31;
    const int wave = tid >> 5;
    const int lo   = lane & 15;   // N / M position within fragment
    const int hi   = lane >> 4;   // K-half (A/B), M+8 half (C/D)

    int row0 = blockIdx.x * ROWS_PER_BLOCK + wave * ROWS_PER_WAVE;
    if (row0 > NROWS - ROWS_PER_WAVE) row0 = NROWS - ROWS_PER_WAVE;  // tail clamp
    float* xw = xl + wave * (ROWS_PER_WAVE * XL_STRIDE);

    // ---- Stage this wave's 16x512 X tile into LDS (b128 stores, padded). ----
    {
        const float4* src = (const float4*)(X + (size_t)row0 * PDIM);
        #pragma unroll 4
        for (int it = 0; it < 64; ++it) {
            int e   = it * 32 + lane;      // float4 index within the tile
            int row = e >> 7;              // 128 float4 per row
            int q   = e & 127;
            float4 v = src[row * (PDIM / 4) + q];
            *(float4*)(xw + row * XL_STRIDE + q * 4) = v;
        }
    }

    // ---- Prologue: kick off step 0's half-tile. ----
    stage_half(smem, ThetaS, tid);

    const float beta0 = beta0p[0];
    float s[8];
    #pragma unroll
    for (int r = 0; r < 8; ++r) s[r] = 0.0f;

    v8f c = {0.f, 0.f, 0.f, 0.f, 0.f, 0.f, 0.f, 0.f};

    for (int st = 0; st < HALF_STEPS; ++st) {
        const int nt = st >> 1;       // column tile
        const int kh = st & 1;        // K-half within the tile
        const int n0 = nt * 16;
        float* cur = smem + (st & 1) * TH_HALF_FLOATS;
        float* nxt = smem + ((st + 1) & 1) * TH_HALF_FLOATS;

        __syncthreads();   // all readers of `nxt` (step st-1) are done

        // Issue step st+1's async loads; they complete under the WMMAs below.
        if (st + 1 < HALF_STEPS)
            stage_half(nxt, ThetaS + (size_t)(st + 1) * TH_HALF_FLOATS, tid);

#if HAVE_ASYNC_LDS
        // Per-wave in-order completion: <=4 outstanding means step-st data
        // (issued one iteration ago) has landed in LDS.
        if (st + 1 < HALF_STEPS)
            asm volatile("s_wait_asynccnt 0x4" ::: "memory");
        else
            asm volatile("s_wait_asynccnt 0x0" ::: "memory");
#endif
        __syncthreads();   // `cur` is ready for every wave

        // ---- 64 WMMAs: accumulate C += X_tile(:,Khalf) @ Theta(Khalf,:) ----
        if (kh == 0) {
            v8f z = {0.f, 0.f, 0.f, 0.f, 0.f, 0.f, 0.f, 0.f};
            c = z;
        }
        #pragma unroll 8
        for (int kc = 0; kc < HK_CHUNKS; ++kc) {
            const int k0 = kc * 4;     // local K offset within the half
            // A (16x4 f32): lane holds row lo, K = kh*256 + k0 + 2*hi + {0,1}
            v2f a = *(const v2f*)(xw + lo * XL_STRIDE + kh * 256 + k0 + 2 * hi);
            // B (4x16 f32): lane holds col lo, same K
            v2f b = *(const v2f*)(cur + (k0 / 2 + hi) * 32 + 2 * lo);
            c = __builtin_amdgcn_wmma_f32_16x16x4_f32(
                    /*neg_a=*/false, a, /*neg_b=*/false, b,
                    /*c_mod=*/(short)0, c, /*reuse_a=*/false, /*reuse_b=*/false);
        }

        // ---- after the second K-half: fold beta, do (C+beta).*X rowsum ----
        if (kh == 1) {
            const float bl = beta[n0 + lo];
            #pragma unroll
            for (int r = 0; r < 8; ++r) {
                // C/D layout: VGPR r holds row m = r + 8*hi, col n = n0 + lo
                float xv = xw[(r + 8 * hi) * XL_STRIDE + n0 + lo];
                s[r] += (c[r] + bl) * xv;
            }
        }
    }

    // ---- butterfly-reduce each row across the 16 lanes of its half ----
    #pragma unroll
    for (int r = 0; r < 8; ++r) {
        float v = s[r];
        v += __shfl_xor(v, 1, 32);
        v += __shfl_xor(v, 2, 32);
        v += __shfl_xor(v, 4, 32);
        v += __shfl_xor(v, 8, 32);
        s[r] = v;
    }
    if (lo == 0) {
        #pragma unroll
        for (int r = 0; r < 8; ++r)
            out[row0 + 8 * hi + r] = beta0 + s[r];
    }
}

// ---------------------------------------------------------------------------
extern "C" void kernel_launch(void* const* d_in, const int* in_sizes, int n_in,
                              void* d_out, int out_size, void* d_ws, size_t ws_size,
                              hipStream_t stream) {
    (void)in_sizes; (void)n_in; (void)out_size; (void)ws_size;
    const float* X      = (const float*)d_in[0];   // (200000, 512)
    const float* beta0  = (const float*)d_in[1];   // (1,)
    const float* beta   = (const float*)d_in[2];   // (512,)
    const float* tu     = (const float*)d_in[3];   // (130816,)
    float*       ThetaS = (float*)d_ws;            // 512*512*4 = 1 MB scratch
    float*       outp   = (float*)d_out;           // (200000,)

    // Build Theta directly in tile-major pair-swizzled layout.
    fim_build_theta<<<(PDIM * PDIM) / 256, 256, 0, stream>>>(tu, ThetaS);

    // Fused GEMM + quadratic-form rowsum.
    size_t shmem = (size_t)(2 * TH_HALF_FLOATS +
                            WAVES_PER_BLOCK * ROWS_PER_WAVE * XL_STRIDE)
                   * sizeof(float);                 // 296,960 B < 320 KB
    int grid = (NROWS + ROWS_PER_BLOCK - 1) / ROWS_PER_BLOCK;   // 1563
    fim_main<<<grid, 256, shmem, stream>>>(X, beta0, beta, ThetaS, outp);
}